// Model_70437463654666
// MI455X (gfx1250) — compile-verified
//
#include <hip/hip_runtime.h>

#define HIDDEN 128
#define TILE_M 16

typedef __attribute__((ext_vector_type(2))) float v2f;
typedef __attribute__((ext_vector_type(8))) float v8f;

// ---------------------------------------------------------------------------
// Zero a float buffer.
// ---------------------------------------------------------------------------
__global__ void gnn_zero_kernel(float* __restrict__ p, size_t n) {
    size_t i = (size_t)blockIdx.x * blockDim.x + threadIdx.x;
    if (i < n) p[i] = 0.0f;
}

// ---------------------------------------------------------------------------
// Scatter-add: one wave32 per edge. Each lane carries a float4 (32*4 = 128).
// agg[dst] += h[src]; optionally deg[dst] += 1 (layer 1 only).
// ---------------------------------------------------------------------------
__global__ __launch_bounds__(256)
void gnn_scatter_kernel(const float* __restrict__ h,
                        const long long* __restrict__ src,
                        const long long* __restrict__ dst,
                        float* __restrict__ agg,
                        float* __restrict__ deg,
                        int n_edges, int add_deg) {
    int e = (int)(((size_t)blockIdx.x * blockDim.x + threadIdx.x) >> 5);
    int lane = threadIdx.x & 31;
    if (e >= n_edges) return;           // uniform per wave

    int s = (int)src[e];
    int d = (int)dst[e];

    const float4* hs = (const float4*)(h + (size_t)s * HIDDEN);
    float4 v = hs[lane];

    float* base = agg + (size_t)d * HIDDEN + lane * 4;
    unsafeAtomicAdd(base + 0, v.x);     // global_atomic_add_f32
    unsafeAtomicAdd(base + 1, v.y);
    unsafeAtomicAdd(base + 2, v.z);
    unsafeAtomicAdd(base + 3, v.w);
    if (add_deg && lane == 0) unsafeAtomicAdd(&deg[d], 1.0f);
}

// ---------------------------------------------------------------------------
// Fused SAGEConv GEMM:
//   out = (agg/max(deg,1)) @ Wl^T + xin @ Wr^T + bl   (+ optional ReLU)
// Block = 256 threads = 8 waves. Block covers 16 node rows; wave w covers
// output columns [16w, 16w+16). V_WMMA_F32_16X16X4_F32, K stepped by 4,
// two GEMMs fused into one accumulator seeded with the bias.
//
// Layouts per CDNA5 ISA §7.12.2 (wave32):
//   A 16x4 f32 : lanes 0-15 -> M=lane, VGPR0=K0, VGPR1=K1
//                lanes 16-31 -> M=lane-16, VGPR0=K2, VGPR1=K3
//   B 4x16 f32 : mirrored K split across lane halves, N = lane&15
//   C/D 16x16  : VGPR r -> M = r + 8*(lane>=16), N = lane&15
//
// D is staged back through LDS (reusing the Amean buffer) so the HBM
// write-out is coalesced b128 stores instead of per-lane strided b32.
// ---------------------------------------------------------------------------
template <bool RELU>
__global__ __launch_bounds__(256)
void gnn_sage_gemm_kernel(const float* __restrict__ agg,   // [N,128] sums
                          const float* __restrict__ deg,   // [N]
                          const float* __restrict__ xin,   // [N,128] root feats
                          const float* __restrict__ Wl,    // [128,128]
                          const float* __restrict__ bl,    // [128]
                          const float* __restrict__ Wr,    // [128,128]
                          float* __restrict__ out,         // [N,128]
                          int n_nodes) {
    __shared__ float Amean[TILE_M][HIDDEN];   // reused as D tile after compute
    __shared__ float Aroot[TILE_M][HIDDEN];

    const int block_row = blockIdx.x * TILE_M;
    const int tid = threadIdx.x;

    // Cooperative stage of 16x128 mean (deg-scaled) + 16x128 root features.
    for (int i = tid; i < TILE_M * HIDDEN; i += 256) {
        int r = i / HIDDEN, c = i % HIDDEN;
        int node = block_row + r;
        float m = 0.0f, xv = 0.0f;
        if (node < n_nodes) {
            float scale = 1.0f / fmaxf(deg[node], 1.0f);
            m  = agg[(size_t)node * HIDDEN + c] * scale;
            xv = xin[(size_t)node * HIDDEN + c];
        }
        Amean[r][c] = m;
        Aroot[r][c] = xv;
    }
    __syncthreads();

    const int wave = tid >> 5;          // 0..7 -> N tile
    const int lane = tid & 31;
    const int half = lane >> 4;         // 0: lanes 0-15, 1: lanes 16-31
    const int m16  = lane & 15;
    const int ncol = wave * 16 + m16;   // output column this lane owns

    v8f acc;
    const float bias = bl[ncol];
#pragma unroll
    for (int r = 0; r < 8; ++r) acc[r] = bias;

#pragma unroll 4
    for (int kb = 0; kb < HIDDEN; kb += 4) {
        const int kk = kb + 2 * half;   // this lane-half's K pair
        // neighbor-mean term: D += Amean * Wl^T
        v2f a1; a1.x = Amean[m16][kk]; a1.y = Amean[m16][kk + 1];
        v2f b1; b1.x = Wl[(size_t)ncol * HIDDEN + kk];
                b1.y = Wl[(size_t)ncol * HIDDEN + kk + 1];
        acc = __builtin_amdgcn_wmma_f32_16x16x4_f32(
            false, a1, false, b1, (short)0, acc, false, false);
        // root term: D += Aroot * Wr^T
        v2f a2; a2.x = Aroot[m16][kk]; a2.y = Aroot[m16][kk + 1];
        v2f b2; b2.x = Wr[(size_t)ncol * HIDDEN + kk];
                b2.y = Wr[(size_t)ncol * HIDDEN + kk + 1];
        acc = __builtin_amdgcn_wmma_f32_16x16x4_f32(
            false, a2, false, b2, (short)0, acc, false, false);
    }

    // Stage D into LDS (reuse Amean) so the global write-out is coalesced.
    __syncthreads();                     // all A reads done before overwrite
#pragma unroll
    for (int r = 0; r < 8; ++r) {
        float v = acc[r];
        if (RELU) v = fmaxf(v, 0.0f);
        Amean[r + 8 * half][ncol] = v;   // row = M within tile, col = N
    }
    __syncthreads();

    // Coalesced write-out: each thread owns 8 consecutive floats (2 x b128).
    {
        const int off = tid * 8;              // 256 threads * 8 = 2048 floats
        const int r   = off / HIDDEN;
        const int c   = off % HIDDEN;
        const int row = block_row + r;
        if (row < n_nodes) {
            const float4* s4 = (const float4*)&Amean[r][c];
            float4* o4 = (float4*)(out + (size_t)row * HIDDEN + c);
            o4[0] = s4[0];
            o4[1] = s4[1];
        }
    }
}

// ---------------------------------------------------------------------------
// Edge scorer on raw features: pred[e] = dot(x[src[e]], x[dst[e]]).
// One wave32 per edge, float4 per lane, shuffle reduction.
// ---------------------------------------------------------------------------
__global__ __launch_bounds__(256)
void gnn_edge_pred_kernel(const float* __restrict__ x,
                          const long long* __restrict__ src,
                          const long long* __restrict__ dst,
                          float* __restrict__ pred, int n_edges) {
    int e = (int)(((size_t)blockIdx.x * blockDim.x + threadIdx.x) >> 5);
    int lane = threadIdx.x & 31;
    if (e >= n_edges) return;

    int s = (int)src[e];
    int d = (int)dst[e];
    const float4* xs = (const float4*)(x + (size_t)s * HIDDEN);
    const float4* xd = (const float4*)(x + (size_t)d * HIDDEN);
    float4 a = xs[lane];
    float4 b = xd[lane];
    float sum = a.x * b.x + a.y * b.y + a.z * b.z + a.w * b.w;
#pragma unroll
    for (int off = 16; off > 0; off >>= 1)
        sum += __shfl_xor(sum, off, 32);
    if (lane == 0) pred[e] = sum;
}

// ---------------------------------------------------------------------------
// Launch: zero -> scatter(x) -> gemm1(relu) -> zero -> scatter(h1) -> gemm2
//         -> edge scorer. Outputs: d_out = [pred (E)] ++ [h (N*128)].
// Workspace: agg (N*128 f32) | deg (N f32) | h1 (N*128 f32)  ~103 MB.
// ---------------------------------------------------------------------------
extern "C" void kernel_launch(void* const* d_in, const int* in_sizes, int n_in,
                              void* d_out, int out_size, void* d_ws, size_t ws_size,
                              hipStream_t stream) {
    const float*     x   = (const float*)d_in[0];
    const long long* ei  = (const long long*)d_in[1];   // int64 [2, E]
    const float*     W1l = (const float*)d_in[2];
    const float*     b1l = (const float*)d_in[3];
    const float*     W1r = (const float*)d_in[4];
    const float*     W2l = (const float*)d_in[5];
    const float*     b2l = (const float*)d_in[6];
    const float*     W2r = (const float*)d_in[7];

    const int n_nodes = in_sizes[0] / HIDDEN;
    const int n_edges = in_sizes[1] / 2;
    const long long* src = ei;
    const long long* dst = ei + n_edges;

    float* pred  = (float*)d_out;            // [E]
    float* h_out = pred + n_edges;           // [N,128]

    float* agg = (float*)d_ws;                          // N*128
    float* deg = agg + (size_t)n_nodes * HIDDEN;        // N
    float* h1  = deg + n_nodes;                         // N*128

    const int edge_blocks = (int)(((size_t)n_edges * 32 + 255) / 256);
    const int gemm_blocks = (n_nodes + TILE_M - 1) / TILE_M;

    // ---- Layer 1 ----
    {
        size_t nz = (size_t)n_nodes * HIDDEN + (size_t)n_nodes;  // agg+deg contiguous
        gnn_zero_kernel<<<(int)((nz + 255) / 256), 256, 0, stream>>>(agg, nz);
    }
    gnn_scatter_kernel<<<edge_blocks, 256, 0, stream>>>(
        x, src, dst, agg, deg, n_edges, 1);
    gnn_sage_gemm_kernel<true><<<gemm_blocks, 256, 0, stream>>>(
        agg, deg, x, W1l, b1l, W1r, h1, n_nodes);

    // ---- Layer 2 (reuse deg; re-zero agg) ----
    {
        size_t nz = (size_t)n_nodes * HIDDEN;
        gnn_zero_kernel<<<(int)((nz + 255) / 256), 256, 0, stream>>>(agg, nz);
    }
    gnn_scatter_kernel<<<edge_blocks, 256, 0, stream>>>(
        h1, src, dst, agg, deg, n_edges, 0);
    gnn_sage_gemm_kernel<false><<<gemm_blocks, 256, 0, stream>>>(
        agg, deg, h1, W2l, b2l, W2r, h_out, n_nodes);

    // ---- Edge dot-product scorer on raw features ----
    gnn_edge_pred_kernel<<<edge_blocks, 256, 0, stream>>>(
        x, src, dst, pred, n_edges);
}